// GraphConvolution_2250562863215
// MI455X (gfx1250) — compile-verified
//
#include <hip/hip_runtime.h>
#include <hip/hip_bf16.h>
#include <math.h>

#define HID 512
#define BATCH 16
#define NN 2048

typedef float v2f __attribute__((ext_vector_type(2)));
typedef float v8f __attribute__((ext_vector_type(8)));

__device__ __forceinline__ float wave_reduce_sum(float x) {
    // wave32 butterfly reduction
    x += __shfl_xor(x, 16, 32);
    x += __shfl_xor(x, 8, 32);
    x += __shfl_xor(x, 4, 32);
    x += __shfl_xor(x, 2, 32);
    x += __shfl_xor(x, 1, 32);
    return x;
}

// Kernel 1: per-row sums of clip(adj,0), +1 for identity, d = rsqrt(sum).
// One wave32 per row of 2048 floats (512 float4 coalesced loads).
__global__ void gcn_rowsum_d(const float* __restrict__ adj, float* __restrict__ d) {
    const int wave = (blockIdx.x * blockDim.x + threadIdx.x) >> 5;   // row id in [0, B*N)
    const int lane = threadIdx.x & 31;
    const float4* row = (const float4*)(adj + (size_t)wave * NN);
    float s = 0.0f;
    #pragma unroll 4
    for (int i = lane; i < NN / 4; i += 32) {
        float4 t = row[i];
        s += fmaxf(t.x, 0.0f) + fmaxf(t.y, 0.0f) + fmaxf(t.z, 0.0f) + fmaxf(t.w, 0.0f);
    }
    s = wave_reduce_sum(s);
    if (lane == 0) d[wave] = rsqrtf(s + 1.0f);
}

// Kernel 2: w0[b,m] = d[b,0] * (clip(adj[b,0,m],0) + (m==0)) * d[b,m]
__global__ void gcn_w0(const float* __restrict__ adj, const float* __restrict__ d,
                       float* __restrict__ w0) {
    const int idx = blockIdx.x * blockDim.x + threadIdx.x;   // b*N + m
    const int b = idx >> 11;              // / NN
    const int m = idx & (NN - 1);
    float a = fmaxf(adj[(size_t)b * NN * NN + m], 0.0f) + (m == 0 ? 1.0f : 0.0f);
    w0[idx] = d[b * NN] * a * d[idx];
}

// Kernel 3: v[b,h] = sum_m w0[b,m] * feat[b,h,m]. One wave32 per (b,h) row.
__global__ void gcn_weighted_feat(const float* __restrict__ feat,
                                  const float* __restrict__ w0,
                                  float* __restrict__ v) {
    const int wave = (blockIdx.x * blockDim.x + threadIdx.x) >> 5;   // b*H + h
    const int lane = threadIdx.x & 31;
    const int b = wave >> 9;                                         // / HID
    const float4* frow = (const float4*)(feat + (size_t)wave * NN);
    const float4* wrow = (const float4*)(w0 + (size_t)b * NN);
    float s = 0.0f;
    #pragma unroll 4
    for (int i = lane; i < NN / 4; i += 32) {
        float4 f = frow[i];
        float4 w = wrow[i];
        s += f.x * w.x + f.y * w.y + f.z * w.z + f.w * w.w;
    }
    s = wave_reduce_sum(s);
    if (lane == 0) v[wave] = s;
}

// Kernel 4: out[b,k] = tanh( sum_h v[b,h]*W[h,k] + bias[b,k] )
// Full-f32 WMMA: V_WMMA_F32_16X16X4_F32, M=16 batches, one wave per 16-wide
// k-tile, K=512 reduced in steps of 4. EXEC all-ones (no divergence).
__global__ void gcn_wmma_out(const float* __restrict__ v,     // [16,512]
                             const float* __restrict__ W,     // [512,512]
                             const float* __restrict__ bias,  // [16,512]
                             float* __restrict__ out) {       // [16,512]
    const int lane = threadIdx.x & 31;
    const int tile = blockIdx.x * (blockDim.x >> 5) + (threadIdx.x >> 5);
    const int n0 = tile * 16;          // output column tile
    const int half = lane >> 4;        // 0: lanes 0-15, 1: lanes 16-31
    const int l15 = lane & 15;

    v8f acc = {};
    for (int kk = 0; kk < HID; kk += 4) {
        const int ka = kk + half * 2;  // this half-wave carries K=ka, ka+1
        v2f a, b;
        // A fragment (16x4, M x K): lane l15 holds row M=l15
        a.x = v[l15 * HID + ka];
        a.y = v[l15 * HID + ka + 1];
        // B fragment (4x16, K x N): row K striped across lanes
        b.x = W[(size_t)ka * HID + n0 + l15];
        b.y = W[(size_t)(ka + 1) * HID + n0 + l15];
        acc = __builtin_amdgcn_wmma_f32_16x16x4_f32(
            /*neg_a=*/false, a, /*neg_b=*/false, b,
            /*c_mod=*/(short)0, acc, /*reuse_a=*/false, /*reuse_b=*/false);
    }

    // C/D layout: VGPR r -> (M=r, lanes 0-15) and (M=r+8, lanes 16-31)
    #pragma unroll
    for (int r = 0; r < 8; ++r) {
        const int m = r + half * 8;
        const int idx = m * HID + n0 + l15;
        out[idx] = tanhf(acc[r] + bias[idx]);
    }
}

extern "C" void kernel_launch(void* const* d_in, const int* in_sizes, int n_in,
                              void* d_out, int out_size, void* d_ws, size_t ws_size,
                              hipStream_t stream) {
    const float* feat = (const float*)d_in[0];   // [16, 512, 2048]
    const float* adj  = (const float*)d_in[1];   // [16, 2048, 2048]
    const float* W    = (const float*)d_in[2];   // [512, 512]
    const float* bias = (const float*)d_in[3];   // [16, 512]
    float* out = (float*)d_out;                  // [16, 512]

    float* d  = (float*)d_ws;                    // [16*2048]
    float* w0 = d + BATCH * NN;                  // [16*2048]
    float* v  = w0 + BATCH * NN;                 // [16*512]

    // 1) d[b,m] = rsqrt(rowsum(clip(adj))+1): 32768 rows, 8 waves/block
    gcn_rowsum_d<<<(BATCH * NN) / 8, 256, 0, stream>>>(adj, d);
    // 2) w0[b,m]
    gcn_w0<<<(BATCH * NN) / 256, 256, 0, stream>>>(adj, d, w0);
    // 3) v[b,h]: 8192 rows, 8 waves/block
    gcn_weighted_feat<<<(BATCH * HID) / 8, 256, 0, stream>>>(feat, w0, v);
    // 4) out = tanh(v @ W + bias): 32 tiles of 16x16, 8 waves/block
    gcn_wmma_out<<<(HID / 16) / 8, 256, 0, stream>>>(v, W, bias, out);
}